// DifferentiableVietorisRips_54434415509943
// MI455X (gfx1250) — compile-verified
//
#include <hip/hip_runtime.h>

typedef __attribute__((ext_vector_type(2))) float v2f;
typedef __attribute__((ext_vector_type(4))) float v4f;
typedef __attribute__((ext_vector_type(8))) float v8f;

#define VR_N 128
#define VR_D 512
#define VR_PAIRS 8128          // C(128,2)
#define VR_TRIPLES 341376      // C(128,3)
#define VR_EPS 32.0f
#define VR_SHARP 10.0f

// ---------------------------------------------------------------------------
// Kernel 0: row squared norms. One wave per row, coalesced float4 loads,
// wave32 xor-shuffle reduction.
// ---------------------------------------------------------------------------
__global__ void vr_sq_kernel(const float* __restrict__ W, float* __restrict__ sq) {
    int wave = (blockIdx.x * blockDim.x + threadIdx.x) >> 5;
    int lane = threadIdx.x & 31;
    if (wave >= VR_N) return;
    const float* row = W + (size_t)wave * VR_D;
    float acc = 0.0f;
#pragma unroll
    for (int kk = 0; kk < VR_D; kk += 128) {
        v4f v = *(const v4f*)(row + kk + lane * 4);
        acc += v.x * v.x + v.y * v.y + v.z * v.z + v.w * v.w;
    }
#pragma unroll
    for (int off = 16; off > 0; off >>= 1) acc += __shfl_xor(acc, off, 32);
    if (lane == 0) sq[wave] = acc;
}

// ---------------------------------------------------------------------------
// Kernel 1: distance table via f32 WMMA. One wave per 16x16 tile of G=W*W^T.
// A layout (16x4 f32): lane m (0..15), half=lane>>4; VGPR pair holds
// K = {2*half, 2*half+1}. B (4x16) mirrors it with N across lanes.
// C layout: element v of lane -> row = v + 8*half, col = lane&15.
// ---------------------------------------------------------------------------
__global__ void vr_dist_kernel(const float* __restrict__ W,
                               const float* __restrict__ sq,
                               float* __restrict__ dist) {
    int tile = blockIdx.x;               // 64 tiles (8x8)
    int row0 = (tile >> 3) * 16;
    int col0 = (tile & 7) * 16;
    int lane = threadIdx.x & 31;
    int half = lane >> 4;
    int m    = lane & 15;
    const float* arow = W + (size_t)(row0 + m) * VR_D;
    const float* brow = W + (size_t)(col0 + m) * VR_D;
    v8f c = {};
#pragma unroll 8
    for (int kb = 0; kb < VR_D / 4; ++kb) {
        int kA = kb * 4 + half * 2;
        v2f a = *(const v2f*)(arow + kA);
        v2f b = *(const v2f*)(brow + kA);
        c = __builtin_amdgcn_wmma_f32_16x16x4_f32(
                /*neg_a=*/false, a, /*neg_b=*/false, b,
                /*c_mod=*/(short)0, c, /*reuse_a=*/false, /*reuse_b=*/false);
    }
    int col = col0 + m;
    float sqc = sq[col];
#pragma unroll
    for (int v = 0; v < 8; ++v) {
        int row = row0 + v + 8 * half;
        float d2 = sq[row] + sqc - 2.0f * c[v];
        dist[row * VR_N + col] = sqrtf(fmaxf(d2, 0.0f));
    }
}

// ---------------------------------------------------------------------------
// Kernel 2: m0 (identity) + m1 (pair rows). One wave per output row; each
// lane writes one float4 (16B) with a non-temporal streaming store.
// ---------------------------------------------------------------------------
__global__ void vr_m01_kernel(const float* __restrict__ dist,
                              float* __restrict__ out) {
    int wave = (blockIdx.x * blockDim.x + threadIdx.x) >> 5;
    int lane = threadIdx.x & 31;
    if (wave >= VR_N + VR_PAIRS) return;
    int c0 = lane * 4;
    v4f o = {0.0f, 0.0f, 0.0f, 0.0f};
    if (wave < VR_N) {
        int r = wave;
        o.x = (c0 + 0 == r) ? 1.0f : 0.0f;
        o.y = (c0 + 1 == r) ? 1.0f : 0.0f;
        o.z = (c0 + 2 == r) ? 1.0f : 0.0f;
        o.w = (c0 + 3 == r) ? 1.0f : 0.0f;
    } else {
        int p = wave - VR_N;
        int i = 0, rem = p;
        while (rem >= (VR_N - 1 - i)) { rem -= (VR_N - 1 - i); ++i; }
        int j = i + 1 + rem;
        float d   = dist[i * VR_N + j];
        float val = 1.0f / (1.0f + __expf(VR_SHARP * (d - VR_EPS)));
        o.x = (c0 + 0 == i || c0 + 0 == j) ? val : 0.0f;
        o.y = (c0 + 1 == i || c0 + 1 == j) ? val : 0.0f;
        o.z = (c0 + 2 == i || c0 + 2 == j) ? val : 0.0f;
        o.w = (c0 + 3 == i || c0 + 3 == j) ? val : 0.0f;
    }
    __builtin_nontemporal_store(o, (v4f*)(out + (size_t)wave * VR_N + c0));
}

// ---------------------------------------------------------------------------
// Kernel 3: m2 (triple rows). 8 consecutive rows per wave: one linear decode
// of t -> (i,j,k), then O(1) incremental updates. NT float4 stores.
// ---------------------------------------------------------------------------
__global__ void vr_m2_kernel(const float* __restrict__ dist,
                             float* __restrict__ out) {
    const int ROWS = 8;
    int  wave = (blockIdx.x * blockDim.x + threadIdx.x) >> 5;
    int  lane = threadIdx.x & 31;
    long t0   = (long)wave * ROWS;
    if (t0 >= VR_TRIPLES) return;

    // decode t0 -> (i,j,k) in combinations(range(128),3) order
    int i = 0;
    long rem = t0;
    for (;;) {
        long cnt = (long)(VR_N - 1 - i) * (VR_N - 2 - i) / 2;  // triples with this i
        if (rem < cnt) break;
        rem -= cnt; ++i;
    }
    int j = i + 1;
    while (rem >= (VR_N - 1 - j)) { rem -= (VR_N - 1 - j); ++j; }
    int k = j + 1 + (int)rem;

    int c0 = lane * 4;
    float* rowp = out + (size_t)(VR_N + VR_PAIRS + t0) * VR_N + c0;
#pragma unroll
    for (int rr = 0; rr < ROWS; ++rr) {
        float dij = dist[i * VR_N + j];
        float djk = dist[j * VR_N + k];
        float dik = dist[i * VR_N + k];
        float val = (dij <= VR_EPS && djk <= VR_EPS && dik <= VR_EPS) ? 1.0f : 0.0f;
        v4f o;
        o.x = (c0 + 0 == i || c0 + 0 == j || c0 + 0 == k) ? val : 0.0f;
        o.y = (c0 + 1 == i || c0 + 1 == j || c0 + 1 == k) ? val : 0.0f;
        o.z = (c0 + 2 == i || c0 + 2 == j || c0 + 2 == k) ? val : 0.0f;
        o.w = (c0 + 3 == i || c0 + 3 == j || c0 + 3 == k) ? val : 0.0f;
        __builtin_nontemporal_store(o, (v4f*)rowp);
        rowp += VR_N;
        // increment triple (i,j,k)
        ++k;
        if (k >= VR_N) {
            ++j;
            if (j >= VR_N - 1) { ++i; j = i + 1; }
            k = j + 1;
        }
    }
}

// ---------------------------------------------------------------------------
extern "C" void kernel_launch(void* const* d_in, const int* in_sizes, int n_in,
                              void* d_out, int out_size, void* d_ws, size_t ws_size,
                              hipStream_t stream) {
    const float* W  = (const float*)d_in[0];
    float* out      = (float*)d_out;
    float* sq       = (float*)d_ws;            // 128 floats
    float* dist     = sq + VR_N;               // 128*128 floats

    // 1) row norms: 128 waves = 16 blocks x 256 threads
    vr_sq_kernel<<<16, 256, 0, stream>>>(W, sq);

    // 2) WMMA distance table: 64 tiles, one wave each
    vr_dist_kernel<<<64, 32, 0, stream>>>(W, sq, dist);

    // 3) m0 + m1: 8256 rows, one wave per row
    int waves01  = VR_N + VR_PAIRS;                       // 8256
    int blocks01 = (waves01 * 32 + 255) / 256;            // 1032
    vr_m01_kernel<<<blocks01, 256, 0, stream>>>(dist, out);

    // 4) m2: 341376 rows, 8 per wave
    int waves2  = VR_TRIPLES / 8;                         // 42672
    int blocks2 = (waves2 * 32 + 255) / 256;              // 5334
    vr_m2_kernel<<<blocks2, 256, 0, stream>>>(dist, out);
}